// LDAMSegmenterV0_34471407518059
// MI455X (gfx1250) — compile-verified
//
#include <hip/hip_runtime.h>

// Problem constants (from reference): B=8, C=21, H=W=512
#define N_PIX   (8 * 512 * 512)   // 2,097,152 pixels
#define NCLS    21
#define HW_SH   18                // H*W = 262144 = 2^18
#define HW_MASK ((1 << HW_SH) - 1)
#define SCALE_S 30.0f
#define MAX_M_  0.5f

#define HIST_BLOCKS 256
#define HIST_T      256
#define HIST_ITEMS  (N_PIX / (HIST_BLOCKS * HIST_T))   // 32

#define MAIN_BLOCKS 1024
#define MAIN_T      256
#define MAIN_ITEMS  (N_PIX / (MAIN_BLOCKS * MAIN_T))   // 8

typedef __attribute__((ext_vector_type(2))) float v2f;
typedef __attribute__((ext_vector_type(8))) float v8f;

// ---------------------------------------------------------------------------
// 256-thread block sum using V_WMMA_F32_16X16X4_F32 as a reduction primitive:
// D = ones(16x4) x B(4x16) + C  => every B element lands in exactly one of 16
// column sums; C-chaining accumulates 4 chunks of 64 values; 4 shuffle-adds
// over the 16 columns give the grand total. Result valid on lane 0.
// ---------------------------------------------------------------------------
__device__ __forceinline__ float block_reduce_256_wmma(float v) {
  __shared__ float sred[256];
  sred[threadIdx.x] = v;
  __syncthreads();
  float total = 0.0f;
  if (threadIdx.x < 32) {          // wave 0 only; full-EXEC within the wave
    const int L = threadIdx.x;
    v8f accm = {};
    v2f ones; ones[0] = 1.0f; ones[1] = 1.0f;
#pragma unroll
    for (int j = 0; j < 4; ++j) {
      v2f bb; bb[0] = sred[j * 64 + L]; bb[1] = sred[j * 64 + 32 + L];
      accm = __builtin_amdgcn_wmma_f32_16x16x4_f32(false, ones, false, bb,
                                                   (short)0, accm, false, false);
    }
    float s = accm[0];             // lanes 0..15 hold the 16 column sums
    s += __shfl_xor(s, 1);
    s += __shfl_xor(s, 2);
    s += __shfl_xor(s, 4);
    s += __shfl_xor(s, 8);
    total = s;                     // lane 0: sum of all 256 inputs
  }
  return total;
}

// ---------------------------------------------------------------------------
// Kernel 1: per-block label histogram (LDS bins, integer-exact, no atomics on
// global memory -> deterministic and no init kernel required).
// ---------------------------------------------------------------------------
__global__ void __launch_bounds__(HIST_T)
ldam_hist_kernel(const int* __restrict__ target, unsigned* __restrict__ blockCounts) {
  __shared__ unsigned h[NCLS];
  if (threadIdx.x < NCLS) h[threadIdx.x] = 0u;
  __syncthreads();
  const int tid = blockIdx.x * HIST_T + threadIdx.x;
  const int stride = HIST_BLOCKS * HIST_T;
#pragma unroll
  for (int it = 0; it < HIST_ITEMS; ++it) {
    const int lbl = target[tid + it * stride];
    atomicAdd(&h[lbl], 1u);
  }
  __syncthreads();
  if (threadIdx.x < NCLS)
    blockCounts[blockIdx.x * NCLS + threadIdx.x] = h[threadIdx.x];
}

// ---------------------------------------------------------------------------
// Kernel 2: reduce block histograms, compute m_list (one wave32).
// m = (cnt + 1e-4)^(-1/4), scaled so max(m) == 0.5.
// ---------------------------------------------------------------------------
__global__ void ldam_mlist_kernel(const unsigned* __restrict__ blockCounts,
                                  float* __restrict__ mlist) {
  const int t = threadIdx.x;       // 32 threads
  float mraw = 0.0f, mmax = -1.0f;
  if (t < NCLS) {
    unsigned tot = 0;
    for (int b = 0; b < HIST_BLOCKS; ++b) tot += blockCounts[b * NCLS + t];
    const float cnt = (float)tot + 1e-4f;
    mraw = 1.0f / sqrtf(sqrtf(cnt));
    mmax = mraw;
  }
#pragma unroll
  for (int off = 16; off >= 1; off >>= 1)
    mmax = fmaxf(mmax, __shfl_xor(mmax, off));
  if (t < NCLS) mlist[t] = mraw * (MAX_M_ / mmax);
}

// ---------------------------------------------------------------------------
// Kernel 3: main streaming pass. Lane-per-pixel, loop over 21 channels gives
// perfectly coalesced 128B/wave nontemporal loads (channel-major layout).
// All 21 scaled logits stay in VGPRs -> single pass over 176 MB of pred.
// Margin table broadcast to LDS via the CDNA5 async global->LDS path.
// ---------------------------------------------------------------------------
__global__ void __launch_bounds__(MAIN_T)
ldam_loss_kernel(const float* __restrict__ pred, const int* __restrict__ target,
                 const float* __restrict__ mlist, float* __restrict__ partials) {
  __shared__ float s_m[32];
  if (threadIdx.x < NCLS) {
    unsigned lds_off = (unsigned)(size_t)(&s_m[threadIdx.x]);           // addr[31:0] == LDS offset
    unsigned long long ga = (unsigned long long)(size_t)(mlist + threadIdx.x);
    asm volatile("global_load_async_to_lds_b32 %0, %1, off"
                 :: "v"(lds_off), "v"(ga) : "memory");
    asm volatile("s_wait_asynccnt 0" ::: "memory");
  }
  __syncthreads();

  const int tid = blockIdx.x * MAIN_T + threadIdx.x;
  const int stride = MAIN_BLOCKS * MAIN_T;                               // 262144 = one image
  float acc = 0.0f;
  for (int it = 0; it < MAIN_ITEMS; ++it) {
    const int p = tid + it * stride;
    const int b = p >> HW_SH;
    const int q = p & HW_MASK;
    const int lbl = target[p];
    const float negsm = -SCALE_S * s_m[lbl];
    const float* base = pred + (((size_t)(b * NCLS)) << HW_SH) + (size_t)q;

    float zv[NCLS];
    float mx = -3.4e38f;
    float zt = 0.0f;
#pragma unroll
    for (int c = 0; c < NCLS; ++c) {
      const float x = __builtin_nontemporal_load(base + ((size_t)c << HW_SH));
      const float z = fmaf(SCALE_S, x, (c == lbl) ? negsm : 0.0f);
      zv[c] = z;
      mx = fmaxf(mx, z);
      if (c == lbl) zt = z;        // capture target logit: no dynamic reg index
    }
    float se = 0.0f;
#pragma unroll
    for (int c = 0; c < NCLS; ++c) se += __expf(zv[c] - mx);
    acc += __logf(se) + mx - zt;   // nll = logsumexp - z_target
  }

  const float bsum = block_reduce_256_wmma(acc);
  if (threadIdx.x == 0) partials[blockIdx.x] = bsum;
}

// ---------------------------------------------------------------------------
// Kernel 4: deterministic final reduction of 1024 partials, divide by N.
// ---------------------------------------------------------------------------
__global__ void __launch_bounds__(256)
ldam_final_kernel(const float* __restrict__ partials, float* __restrict__ out) {
  float acc = 0.0f;
#pragma unroll
  for (int k = 0; k < MAIN_BLOCKS / 256; ++k)
    acc += partials[threadIdx.x + k * 256];
  const float tot = block_reduce_256_wmma(acc);
  if (threadIdx.x == 0) out[0] = tot / (float)N_PIX;
}

// ---------------------------------------------------------------------------
// Workspace layout (all uint32/float slots, ~26 KB total):
//   [0 .. 5376)          : unsigned blockCounts[HIST_BLOCKS * NCLS]
//   [5376 .. 5408)       : float mlist[21] (padded to 32)
//   [5408 .. 6432)       : float partials[MAIN_BLOCKS]
// Every slot that is read is written earlier in the same launch -> no
// dependence on poisoned workspace contents; fully deterministic.
// ---------------------------------------------------------------------------
extern "C" void kernel_launch(void* const* d_in, const int* in_sizes, int n_in,
                              void* d_out, int out_size, void* d_ws, size_t ws_size,
                              hipStream_t stream) {
  const float* pred  = (const float*)d_in[0];
  const int* target  = (const int*)d_in[1];
  unsigned* blockCounts = (unsigned*)d_ws;
  float* mlist    = (float*)d_ws + HIST_BLOCKS * NCLS;
  float* partials = (float*)d_ws + HIST_BLOCKS * NCLS + 32;
  float* out = (float*)d_out;

  ldam_hist_kernel<<<HIST_BLOCKS, HIST_T, 0, stream>>>(target, blockCounts);
  ldam_mlist_kernel<<<1, 32, 0, stream>>>(blockCounts, mlist);
  ldam_loss_kernel<<<MAIN_BLOCKS, MAIN_T, 0, stream>>>(pred, target, mlist, partials);
  ldam_final_kernel<<<1, 256, 0, stream>>>(partials, out);
}